// OrthogonalBlock_61607010894026
// MI455X (gfx1250) — compile-verified
//
#include <hip/hip_runtime.h>
#include <math.h>

// Problem shape (reference: N=1048576 rows, M=64 cols, fp32, row-major).
#define MCOLS 64

typedef float v2f __attribute__((ext_vector_type(2)));
typedef float v8f __attribute__((ext_vector_type(8)));

// ---------------------------------------------------------------------------
// Kernel 0: zero the 64x64 Gram accumulator in workspace.
// ---------------------------------------------------------------------------
__global__ void ob_zero_kernel(float* __restrict__ G) {
    int i = blockIdx.x * blockDim.x + threadIdx.x;
    if (i < MCOLS * MCOLS) G[i] = 0.0f;
}

// ---------------------------------------------------------------------------
// Kernel 1: G = X^T X via V_WMMA_F32_16X16X4_F32.
//
// Tile grid is 4x4 (64x64 Gram in 16x16 tiles). Each wave owns ONE tile-row
// `a` (4 accumulator tiles = 32 VGPRs, no spills) and computes
//     acc[b] += Afrag * frag[b],  b = 0..3
// The A-fragment equals frag[a] but is loaded directly from its own
// wave-uniform column offset (L0 hit) instead of indexing frag[] with a
// runtime value -- avoids a v_cndmask select chain in the hot loop.
// The 8 waves of a block form 2 groups of 4 waves; all 4 waves of a group
// stream the same 4-row chunks of X (redundant fragment loads are served by
// the WGP cache, so HBM sees one pass over X). Two K-chunks (8 rows) are
// processed per loop trip. Block-level LDS reduction (ds_add_f32), then one
// set of global float atomics per block.
// ---------------------------------------------------------------------------
__global__ void __launch_bounds__(256) ob_gram_kernel(const float* __restrict__ X,
                                                      float* __restrict__ G,
                                                      int nrows) {
    __shared__ float sG[MCOLS * MCOLS];          // 16 KB block accumulator

    const int lane = threadIdx.x & 31;
    const int m    = lane & 15;                  // M (and N) index within tile
    const int kb   = (lane >> 4) << 1;           // K base within chunk: 0 or 2
    const int wib  = threadIdx.x >> 5;           // wave in block, 0..7
    const int a    = wib & 3;                    // tile-row owned by this wave
    const int grp  = wib >> 2;                   // chunk group, 0..1
    const int ca   = a * 16 + m;                 // column offset of A-fragment

    const long group   = (long)blockIdx.x * 2 + grp;
    const long ngroups = (long)gridDim.x * 2;

    // zero LDS accumulator
    for (int i = threadIdx.x; i < MCOLS * MCOLS; i += blockDim.x) sG[i] = 0.0f;
    __syncthreads();

    v8f acc[4];
    {
        v8f z = {};
        #pragma unroll
        for (int b = 0; b < 4; ++b) acc[b] = z;
    }

    // Each group-iteration covers 8 rows: two 4-row K-chunks.
    for (long r0 = group * 8; r0 + 7 < (long)nrows; r0 += ngroups * 8) {
        // frag0[cg]: X[r0  ..r0+3, 16cg..16cg+15] in WMMA layout
        // frag1[cg]: X[r0+4..r0+7, 16cg..16cg+15]
        const float* b00 = X + (r0 + kb) * MCOLS;      // chunk0, row kb
        const float* b01 = b00 + MCOLS;                // chunk0, row kb+1
        const float* b10 = b00 + 4 * MCOLS;            // chunk1, row kb
        const float* b11 = b10 + MCOLS;                // chunk1, row kb+1
        v2f frag0[4], frag1[4];
        #pragma unroll
        for (int cg = 0; cg < 4; ++cg) {
            frag0[cg][0] = b00[cg * 16 + m];
            frag0[cg][1] = b01[cg * 16 + m];
            frag1[cg][0] = b10[cg * 16 + m];
            frag1[cg][1] = b11[cg * 16 + m];
        }
        // A-fragments: same data as frag0[a]/frag1[a], loaded directly.
        v2f af0, af1;
        af0[0] = b00[ca];  af0[1] = b01[ca];
        af1[0] = b10[ca];  af1[1] = b11[ca];

        #pragma unroll
        for (int b = 0; b < 4; ++b) {
            acc[b] = __builtin_amdgcn_wmma_f32_16x16x4_f32(
                false, af0, false, frag0[b], (short)0, acc[b], false, false);
            acc[b] = __builtin_amdgcn_wmma_f32_16x16x4_f32(
                false, af1, false, frag1[b], (short)0, acc[b], false, false);
        }
    }

    // C/D layout: VGPR g -> row (g | g+8) for lanes 0-15 / 16-31; col = lane&15
    const int rofs = (lane >> 4) ? 8 : 0;
    #pragma unroll
    for (int b = 0; b < 4; ++b)
        #pragma unroll
        for (int g = 0; g < 8; ++g) {
            int row = a * 16 + g + rofs;
            int col = b * 16 + m;
            atomicAdd(&sG[row * MCOLS + col], acc[b][g]);          // ds_add_f32
        }
    __syncthreads();

    for (int i = threadIdx.x; i < MCOLS * MCOLS; i += blockDim.x)
        atomicAdd(&G[i], sG[i]);                                   // global f32 atomic
}

// ---------------------------------------------------------------------------
// Kernel 2: Cholesky G = R^T R (R upper-triangular) and Rinv = R^{-1}.
// 64x64 — one workgroup of 64 threads; cost is negligible vs the sweeps.
// ---------------------------------------------------------------------------
__global__ void ob_factor_kernel(const float* __restrict__ G,
                                 float* __restrict__ Rinv) {
    __shared__ float A[MCOLS][MCOLS + 1];
    const int t = threadIdx.x;   // 0..63, owns column t

    for (int i = 0; i < MCOLS; ++i) A[i][t] = G[i * MCOLS + t];
    __syncthreads();

    // Right-looking Cholesky, upper-triangular result in A[i][j], i<=j.
    for (int k = 0; k < MCOLS; ++k) {
        if (t == k) A[k][k] = sqrtf(A[k][k]);
        __syncthreads();
        if (t > k) A[k][t] /= A[k][k];
        __syncthreads();
        for (int i = k + 1; i <= t; ++i)
            A[i][t] -= A[k][i] * A[k][t];
        __syncthreads();
    }

    // Back-substitute R x = e_t  (thread t builds column t of Rinv).
    float x[MCOLS];
    x[t] = 1.0f / A[t][t];
    for (int i = t - 1; i >= 0; --i) {
        float s = 0.0f;
        for (int j = i + 1; j <= t; ++j) s += A[i][j] * x[j];
        x[i] = -s / A[i][i];
    }
    for (int i = 0; i < MCOLS; ++i)
        Rinv[i * MCOLS + t] = (i <= t) ? x[i] : 0.0f;
}

// ---------------------------------------------------------------------------
// Kernel 3: Q = X * Rinv via V_WMMA_F32_16X16X4_F32.
// One wave per 16-row slab; 4 output tiles (64 cols); K swept 4 at a time.
// A-frag is a contiguous float2 per lane (b64 load); B-frags come from the
// 16 KB Rinv which stays resident in cache.
// ---------------------------------------------------------------------------
__global__ void __launch_bounds__(256) ob_apply_kernel(const float* __restrict__ X,
                                                       const float* __restrict__ Rinv,
                                                       float* __restrict__ Q,
                                                       int nrows) {
    const int  lane = threadIdx.x & 31;
    const int  m    = lane & 15;
    const int  kb   = (lane >> 4) << 1;          // 0 or 2
    const long slab = (long)blockIdx.x * (blockDim.x >> 5) + (threadIdx.x >> 5);
    const long row0 = slab * 16;
    if (row0 >= (long)nrows) return;

    v8f acc[4];
    {
        v8f z = {};
        #pragma unroll
        for (int n = 0; n < 4; ++n) acc[n] = z;
    }

    const float* rowPtr = X + (row0 + m) * MCOLS + kb;   // 8B-aligned
    #pragma unroll
    for (int k0 = 0; k0 < MCOLS; k0 += 4) {
        v2f av = *(const v2f*)(rowPtr + k0);             // X[row0+m][k0+kb .. +kb+1]
        #pragma unroll
        for (int n = 0; n < 4; ++n) {
            v2f bv;
            bv[0] = Rinv[(k0 + kb)     * MCOLS + n * 16 + m];
            bv[1] = Rinv[(k0 + kb + 1) * MCOLS + n * 16 + m];
            acc[n] = __builtin_amdgcn_wmma_f32_16x16x4_f32(
                false, av, false, bv, (short)0, acc[n], false, false);
        }
    }

    const int rofs = (lane >> 4) ? 8 : 0;
    #pragma unroll
    for (int n = 0; n < 4; ++n)
        #pragma unroll
        for (int g = 0; g < 8; ++g)
            Q[(row0 + g + rofs) * MCOLS + n * 16 + m] = acc[n][g];
}

// ---------------------------------------------------------------------------
// Host launcher
// ---------------------------------------------------------------------------
extern "C" void kernel_launch(void* const* d_in, const int* in_sizes, int n_in,
                              void* d_out, int out_size, void* d_ws, size_t ws_size,
                              hipStream_t stream) {
    (void)n_in; (void)out_size; (void)ws_size;
    const float* X = (const float*)d_in[0];
    float*       Q = (float*)d_out;
    float*       G = (float*)d_ws;                 // 64*64 floats
    float*    Rinv = G + MCOLS * MCOLS;            // next 64*64 floats
    const int nrows = in_sizes[0] / MCOLS;

    ob_zero_kernel<<<(MCOLS * MCOLS + 255) / 256, 256, 0, stream>>>(G);
    ob_gram_kernel<<<256, 256, 0, stream>>>(X, G, nrows);
    ob_factor_kernel<<<1, MCOLS, 0, stream>>>(G, Rinv);

    const long slabs  = (long)(nrows + 15) / 16;   // one wave per 16 rows
    const int  wavesPerBlock = 256 / 32;
    const int  blocks = (int)((slabs + wavesPerBlock - 1) / wavesPerBlock);
    ob_apply_kernel<<<blocks, 256, 0, stream>>>(X, Rinv, Q, nrows);
}